// qGPS_32375463477532
// MI455X (gfx1250) — compile-verified
//
#include <hip/hip_runtime.h>

// Problem constants (from reference): B=4096, L=256, LOCAL_DIM=2, M=64
#define B_TOT  4096
#define L_TOT  256
#define M_TOT  64
#define TB     64            // batch rows per block
#define LC     64            // l-chunk length for samples staging
#define NCHUNK (L_TOT / LC)  // 4

typedef __attribute__((ext_vector_type(4))) unsigned int u32x4_t;
typedef __attribute__((ext_vector_type(8))) int          i32x8_t;
typedef __attribute__((ext_vector_type(4))) int          i32x4_t;
typedef __attribute__((ext_vector_type(2))) float        v2f;
typedef __attribute__((ext_vector_type(8))) float        v8f;
typedef __attribute__((ext_vector_type(4))) float        f32x4_t;

// Dynamic LDS map (no static __shared__, so dynamic base = LDS offset 0):
//   [0, 131072)        epsilon, row-major [(s*64+m)][l], 128 rows x 256 f32
//   [131072, +16384)   samples chunk buffer 0, [64 b][64 l] i32  (reused as P[64][64] f32)
//   [147456, +16384)   samples chunk buffer 1
#define EPS_OFF   0u
#define SMP_OFF   131072u
#define SMP_BYTES 16384u
#define LDS_BYTES (131072u + 2u * 16384u)

// ---- Tensor Data Mover: 2D tile of 4-byte elements, global -> LDS ----------
// D# packing per CDNA5 ISA (group0) / §8.4 (group1).
// 6-arg builtin variant: (u32x4 g0, i32x8 g1, i32x4 g2, i32x4 g3, i32x8, i32 cpol)
__device__ __forceinline__ void tdm_load_2d_b32(
    unsigned lds_off, const void* gptr,
    unsigned tile_rows, unsigned tile_cols,
    unsigned tensor_rows, unsigned tensor_cols, unsigned row_pitch_elems)
{
  unsigned long long ga = (unsigned long long)gptr;
  u32x4_t g0;
  g0[0] = 1u;                                      // count=1, user descriptor
  g0[1] = lds_off;                                 // lds_addr (bytes)
  g0[2] = (unsigned)ga;                            // global_addr[31:0]
  g0[3] = (unsigned)((ga >> 32) & 0x01FFFFFFull)   // global_addr[56:32]
        | (2u << 30);                              // type = 2 ("image")
  i32x8_t g1;
  g1[0] = (int)(2u << 16);                         // data_size = 2 -> 4 bytes; mask=0
  g1[1] = (int)((tensor_cols & 0xFFFFu) << 16);    // tensor_dim0[15:0]  (bits 63:48)
  g1[2] = (int)(((tensor_cols >> 16) & 0xFFFFu)    // tensor_dim0[31:16] (bits 79:64)
        | ((tensor_rows & 0xFFFFu) << 16));        // tensor_dim1[15:0]  (bits 95:80)
  g1[3] = (int)(((tensor_rows >> 16) & 0xFFFFu)    // tensor_dim1[31:16]
        | ((tile_cols & 0xFFFFu) << 16));          // tile_dim0 (bits 127:112)
  g1[4] = (int)(tile_rows & 0xFFFFu);              // tile_dim1; tile_dim2 = 0
  g1[5] = (int)row_pitch_elems;                    // tensor_dim0_stride[31:0]
  g1[6] = 0;                                       // stride hi / dim1_stride lo
  g1[7] = 0;
  i32x4_t z4; z4[0] = 0; z4[1] = 0; z4[2] = 0; z4[3] = 0;
  i32x8_t z8;
  z8[0] = 0; z8[1] = 0; z8[2] = 0; z8[3] = 0;
  z8[4] = 0; z8[5] = 0; z8[6] = 0; z8[7] = 0;
  __builtin_amdgcn_tensor_load_to_lds(g0, g1, z4, z4, z8, 0);
}

__global__ void __launch_bounds__(256, 1)
qgps_kernel(const int* __restrict__ samples,
            const float* __restrict__ eps,
            float* __restrict__ out)
{
  extern __shared__ char smem[];
  float* s_eps = (float*)(smem + EPS_OFF);
  int*   s_smp = (int*)(smem + SMP_OFF);

  const int tid  = threadIdx.x;
  const int lane = tid & 31;
  const int wave = tid >> 5;
  const int bb   = blockIdx.x * TB;

  // Wave 0 drives the TDM (EXEC-independent; one issue per wave).
  if (wave == 0) {
    // Whole epsilon tensor: 128 rows x 256 f32, pitch 256 -> LDS [row][l]
    tdm_load_2d_b32(EPS_OFF, eps, 2 * M_TOT, L_TOT, 2 * M_TOT, L_TOT, L_TOT);
    // Samples chunk 0: 64 rows x 64 i32, pitch 256
    tdm_load_2d_b32(SMP_OFF, samples + bb * L_TOT, TB, LC, TB, L_TOT, L_TOT);
  }

  const int bl = tid >> 2;          // 0..63: batch row within block
  const int gm = (tid & 3) << 4;    // m base: 0,16,32,48

  float acc[16];
  #pragma unroll
  for (int j = 0; j < 16; ++j) acc[j] = 1.0f;

  for (int c = 0; c < NCHUNK; ++c) {
    if (wave == 0) {
      if (c + 1 < NCHUNK) {
        // Prefetch next samples chunk into the other buffer, then wait for
        // everything except that in-flight load (TDM completes in order).
        tdm_load_2d_b32(SMP_OFF + (unsigned)((c + 1) & 1) * SMP_BYTES,
                        samples + (bb * L_TOT + (c + 1) * LC),
                        TB, LC, TB, L_TOT, L_TOT);
        __builtin_amdgcn_s_wait_tensorcnt((short)1);
      } else {
        __builtin_amdgcn_s_wait_tensorcnt((short)0);
      }
    }
    __syncthreads();

    const int* smp  = s_smp + (c & 1) * (int)(SMP_BYTES / 4) + bl * LC;
    const int lbase = c * LC;
    for (int lc = 0; lc < LC; ++lc) {
      const int s = smp[lc];                           // 0 or 1
      const float* e = s_eps + ((((s << 6) + gm) << 8) + lbase + lc);
      #pragma unroll
      for (int j = 0; j < 16; ++j) acc[j] *= e[j << 8];  // row stride = 256
    }
    __syncthreads();
  }

  // Stage products P[64 b][64 m] into LDS (reuse samples buffer 0).
  float* s_P = (float*)(smem + SMP_OFF);
  f32x4_t* prow = (f32x4_t*)(s_P + bl * M_TOT + gm);
  #pragma unroll
  for (int q = 0; q < 4; ++q) {
    f32x4_t v;
    v[0] = acc[4 * q + 0]; v[1] = acc[4 * q + 1];
    v[2] = acc[4 * q + 2]; v[3] = acc[4 * q + 3];
    prow[q] = v;
  }
  __syncthreads();

  // Exact fp32 M-reduction with V_WMMA_F32_16X16X4_F32, B = ones(4x16):
  // D[r][*] = sum_m P[r][m]. Waves 0..3 each own a 16-row tile (EXEC all-1s).
  if (wave < 4) {
    const int rbase = wave << 4;
    const int r  = lane & 15;
    const int kk = (lane >> 4) << 1;                  // lanes 0-15: K=0,1; 16-31: K=2,3
    const float* Pr = s_P + (rbase + r) * M_TOT;
    v8f cacc = {};
    v2f onesb; onesb[0] = 1.0f; onesb[1] = 1.0f;
    #pragma unroll
    for (int k0 = 0; k0 < M_TOT; k0 += 4) {
      v2f a; a[0] = Pr[k0 + kk]; a[1] = Pr[k0 + kk + 1];
      cacc = __builtin_amdgcn_wmma_f32_16x16x4_f32(
                 false, a, false, onesb, (short)0, cacc, false, false);
    }
    // D layout: VGPR v, lanes 0-15 -> row v, lanes 16-31 -> row v+8. Col 0 suffices.
    if ((lane & 15) == 0) {
      const int ro = rbase + ((lane >> 4) << 3);
      #pragma unroll
      for (int v = 0; v < 8; ++v) out[bb + ro + v] = cacc[v];
    }
  }
}

extern "C" void kernel_launch(void* const* d_in, const int* in_sizes, int n_in,
                              void* d_out, int out_size, void* d_ws, size_t ws_size,
                              hipStream_t stream) {
  const int*   samples = (const int*)d_in[0];   // (4096, 256) int32
  const float* eps     = (const float*)d_in[1]; // (2, 64, 256) float32
  float*       out     = (float*)d_out;         // (4096,) float32
  (void)in_sizes; (void)n_in; (void)out_size; (void)d_ws; (void)ws_size;

  // 160 KB dynamic LDS (> default 64 KB cap) — raise the limit; idempotent.
  (void)hipFuncSetAttribute((const void*)qgps_kernel,
                            hipFuncAttributeMaxDynamicSharedMemorySize,
                            (int)LDS_BYTES);

  qgps_kernel<<<B_TOT / TB, 256, LDS_BYTES, stream>>>(samples, eps, out);
}